// CiderFeatures_18476949307778
// MI455X (gfx1250) — compile-verified
//
#include <hip/hip_runtime.h>

typedef __attribute__((ext_vector_type(2))) float v2f;
typedef __attribute__((ext_vector_type(8))) float v8f;

#define LOG2E 1.44269504088896340736f
// C2 = (6*pi^2)^(2/3) * (12/(160*pi));  CTTF = 0.3*(3*pi^2)^(2/3)
#define CONST_C2   0.36269811f
#define CONST_CTTF 2.8712340f

#if __has_builtin(__builtin_amdgcn_exp2f)
#define EXP2F(x) __builtin_amdgcn_exp2f(x)
#else
#define EXP2F(x) exp2f(x)
#endif

// ---------------------------------------------------------------------------
// Key identity (from the reference constants, D/A == 1 exactly):
//   beta_i = scale_i * (2 + C2*x_i)
//   a_j        == beta_j
//   b_{i,1..3} == { beta_i/2, beta_i, 2*beta_i }
// => channel exponentials exp(-(beta_j + m*beta_i)*sqd), m in {1/2, 1, 2},
//    reconstructed from TWO exps:  G = exp(-(beta_j + beta_i/2)*sqd),
//                                  H = exp(-(beta_i/2)*sqd)
//    ch1 = G, ch2 = G*H, ch3 = (G*H)*(H*H).
//
// Pass 1 precomputes per point:
//   af[i] = { bl_i = -log2e * beta_i ,  f_i = w_i * rho_i }
//   pA[i] = { x, y, z, r2 }         (WMMA A operand + C diagonal)
//   pB[i] = { -2x, -2y, -2z, r2 }   (WMMA B operand: |xi-xj|^2 via rank-4 GEMM)
// ---------------------------------------------------------------------------
__global__ __launch_bounds__(256) void cider_prep(
    const float* __restrict__ rho, const float* __restrict__ gamma,
    const float* __restrict__ coords, const float* __restrict__ weights,
    float4* __restrict__ pA, float4* __restrict__ pB,
    float2* __restrict__ af, int n)
{
    int i = blockIdx.x * blockDim.x + threadIdx.x;
    if (i >= n) return;

    float r  = rho[i];
    float rp = r + 1e-8f;
    float g  = gamma[i];

    float tw   = g / (8.0f * rp);
    float rp23 = powf(rp, 2.0f / 3.0f);
    float ttf  = CONST_CTTF * rp23 * rp;          // 0.3*(3pi^2)^(2/3) * rp^(5/3)
    float x    = tw / ttf;
    float scale = 3.14159265358979f * powf(0.5f * rp, 2.0f / 3.0f);

    float beta = scale * (2.0f + CONST_C2 * x);
    af[i] = make_float2(-LOG2E * beta, weights[i] * r);

    float cx = coords[3 * i + 0], cy = coords[3 * i + 1], cz = coords[3 * i + 2];
    float r2 = cx * cx + cy * cy + cz * cz;
    pA[i] = make_float4(cx, cy, cz, r2);
    pB[i] = make_float4(-2.0f * cx, -2.0f * cy, -2.0f * cz, r2);
}

// ---------------------------------------------------------------------------
// Pass 2: each wave owns one 16-row i-tile and one j-split.
// sqd tile (16x16) via V_WMMA_F32_16X16X4_F32:
//   A row i = [x_i, y_i, z_i, 1]        (lanes 0-15: K0,K1 ; lanes 16-31: K2,K3)
//   B col j = [-2x_j,-2y_j,-2z_j,r2_j]  (same half-wave K split)
//   C[r]    = r2 of row (rbase+r)
// D layout: VGPR r -> row r (lanes 0-15) / row r+8 (lanes 16-31), col = lane&15.
// ---------------------------------------------------------------------------
__global__ __launch_bounds__(256) void cider_main(
    const float4* __restrict__ pA, const float4* __restrict__ pB,
    const float2* __restrict__ af,
    float* __restrict__ partial, int n, int split)
{
    const int lane = threadIdx.x & 31;
    const int wave = (blockIdx.x * blockDim.x + threadIdx.x) >> 5;
    const int s     = wave % split;
    const int iTile = wave / split;
    const int i0    = iTile * 16;
    if (i0 >= n) return;                       // wave-uniform: EXEC stays full

    const int  col = lane & 15;
    const bool hi  = lane >= 16;

    // A operand (constant over the whole j loop)
    float4 pa = pA[i0 + col];
    v2f Aop;
    Aop.x = hi ? pa.z : pa.x;
    Aop.y = hi ? 1.0f : pa.y;

    // per-row data for this lane's 8 accumulator slots
    const int rbase = i0 + (hi ? 8 : 0);
    v8f   Cop;
    float hrow[8];                             // 0.5 * (-log2e * beta_row)
#pragma unroll
    for (int r = 0; r < 8; ++r) {
        Cop[r]  = pA[rbase + r].w;             // |x_i|^2 completes the distance
        hrow[r] = 0.5f * af[rbase + r].x;
    }

    float acc[8][3];
#pragma unroll
    for (int r = 0; r < 8; ++r)
        for (int c = 0; c < 3; ++c) acc[r][c] = 0.0f;

    const int jchunk = n / split;
    const int jb = s * jchunk;
    for (int j0 = jb; j0 < jb + jchunk; j0 += 16) {
        const int j = j0 + col;
        float4 pb = pB[j];
        v2f Bop;
        Bop.x = hi ? pb.z : pb.x;
        Bop.y = hi ? pb.w : pb.y;
        float2 a_f = af[j];                    // { -log2e*beta_j , f_j } for column j

        v8f D = __builtin_amdgcn_wmma_f32_16x16x4_f32(
            false, Aop, false, Bop, (short)0, Cop, false, false);

#pragma unroll
        for (int r = 0; r < 8; ++r) {
            float t = D[r];                    // sqd(row, col)
            float p = hrow[r] * t;             // -log2e * (beta_i/2) * sqd
            float q = __builtin_fmaf(a_f.x, t, p);
            float G  = EXP2F(q);               // exp(-(beta_j + beta_i/2) sqd)
            float H  = EXP2F(p);               // exp(-(beta_i/2) sqd)
            float GH = G * H;
            float H2 = H * H;
            acc[r][0] = __builtin_fmaf(a_f.y, G,       acc[r][0]);
            acc[r][1] = __builtin_fmaf(a_f.y, GH,      acc[r][1]);
            acc[r][2] = __builtin_fmaf(a_f.y, GH * H2, acc[r][2]);
        }
    }

    // butterfly-reduce over the 16 columns held in each half-wave
#pragma unroll
    for (int m = 8; m >= 1; m >>= 1) {
#pragma unroll
        for (int r = 0; r < 8; ++r)
            for (int c = 0; c < 3; ++c)
                acc[r][c] += __shfl_xor(acc[r][c], m, 32);  // mask<16: stays in half
    }

    if (col == 0) {                            // lane 0 -> rows 0-7, lane 16 -> rows 8-15
        float* pp = partial + (size_t)s * n * 3;
#pragma unroll
        for (int r = 0; r < 8; ++r)
            for (int c = 0; c < 3; ++c)
                pp[3 * (rbase + r) + c] = acc[r][c];
    }
}

// ---------------------------------------------------------------------------
// Pass 3: combine j-splits, apply norms = {1.5^1.5, 2^1.5, 3^1.5}
// ---------------------------------------------------------------------------
__global__ __launch_bounds__(256) void cider_reduce(
    const float* __restrict__ partial, float* __restrict__ out, int n, int split)
{
    int k = blockIdx.x * blockDim.x + threadIdx.x;
    if (k >= n * 3) return;
    float sum = 0.0f;
    for (int s = 0; s < split; ++s) sum += partial[(size_t)s * n * 3 + k];
    const float norms[3] = {1.8371173070873836f, 2.8284271247461903f, 5.196152422706632f};
    out[k] = sum * norms[k % 3];
}

extern "C" void kernel_launch(void* const* d_in, const int* in_sizes, int n_in,
                              void* d_out, int out_size, void* d_ws, size_t ws_size,
                              hipStream_t stream)
{
    const float* rho     = (const float*)d_in[0];
    const float* gamma   = (const float*)d_in[1];
    const float* coords  = (const float*)d_in[2];
    const float* weights = (const float*)d_in[3];
    const int n = in_sizes[0];

    // workspace layout (bytes): pA n*16 | pB n*16 | af n*8 | partial split*n*12
    char* ws = (char*)d_ws;
    float4* pA = (float4*)(ws);
    float4* pB = (float4*)(ws + (size_t)n * 16);
    float2* af = (float2*)(ws + (size_t)n * 32);
    float*  partial = (float*)(ws + (size_t)n * 40);

    // pick j-split (more waves for occupancy) that fits in ws; deterministic.
    int split = 8;
    while (split > 1 && (size_t)n * 40 + (size_t)split * n * 12 > ws_size) split >>= 1;

    cider_prep<<<(n + 255) / 256, 256, 0, stream>>>(rho, gamma, coords, weights,
                                                    pA, pB, af, n);

    const int waves  = ((n + 15) / 16) * split;
    const int blocks = (waves + 7) / 8;        // 256 threads = 8 waves/block
    cider_main<<<blocks, 256, 0, stream>>>(pA, pB, af, partial, n, split);

    cider_reduce<<<(n * 3 + 255) / 256, 256, 0, stream>>>(partial, (float*)d_out, n, split);
}